// GAT4bnE_62740882260463
// MI455X (gfx1250) — compile-verified
//
#include <hip/hip_runtime.h>
#include <hip/hip_bf16.h>
#include <cstdint>
#include <cstddef>

// ---------------------------------------------------------------------------
// GATv2 GNN pipeline for gfx1250 (MI455X), fp32 WMMA (V_WMMA_F32_16X16X4_F32)
// ---------------------------------------------------------------------------

typedef float v2f __attribute__((ext_vector_type(2)));
typedef float v8f __attribute__((ext_vector_type(8)));

__device__ __forceinline__ v8f wmma_f32(v2f a, v2f b, v8f c) {
  // 8 args: (neg_a, A, neg_b, B, c_mod, C, reuse_a, reuse_b)
  return __builtin_amdgcn_wmma_f32_16x16x4_f32(false, a, false, b, (short)0, c,
                                               false, false);
}

// order-preserving uint encoding of float for atomicMax-based segment max
__device__ __forceinline__ unsigned enc_ord(float f) {
  unsigned u = __float_as_uint(f);
  return (u & 0x80000000u) ? ~u : (u | 0x80000000u);
}
__device__ __forceinline__ float dec_ord(unsigned u) {
  unsigned b = (u & 0x80000000u) ? (u & 0x7fffffffu) : ~u;
  return __uint_as_float(b);
}

// ---------------------------------------------------------------------------
// Generic tiled GEMM: C[M,N] = act(A[M,K] @ B[K,N] + bias)
// 8 waves / block; each wave owns one 16x16 C tile (waves tile along N).
// A-frag: lane m=lane&15 holds row m; vgpr0=K(k0+2*half), vgpr1=K(k0+2*half+1)
// ---------------------------------------------------------------------------
template <int ACT>
__global__ __launch_bounds__(256) void gemm_wmma_k(
    const float* __restrict__ A, const float* __restrict__ B,
    const float* __restrict__ bias, float* __restrict__ C, int M, int K,
    int N) {
  const int lane = threadIdx.x & 31;
  const int wv = threadIdx.x >> 5;
  const int tn = blockIdx.y * 8 + wv;
  if (tn * 16 >= N) return;                 // wave-uniform
  const int tm = blockIdx.x;
  const int half = lane >> 4;
  const int m = lane & 15;
  const int n = tn * 16 + m;
  const long row = (long)tm * 16 + m;

  v8f acc = {0.f, 0.f, 0.f, 0.f, 0.f, 0.f, 0.f, 0.f};
  const float* Arow = A + row * (long)K;
  for (int k0 = 0; k0 < K; k0 += 4) {
    v2f a = *(const v2f*)(Arow + k0 + 2 * half);
    v2f b;
    b.x = B[(long)(k0 + 2 * half) * N + n];
    b.y = B[(long)(k0 + 2 * half + 1) * N + n];
    acc = wmma_f32(a, b, acc);
  }
  const float bv = bias ? bias[n] : 0.f;
#pragma unroll
  for (int r = 0; r < 8; ++r) {
    long rr = (long)tm * 16 + half * 8 + r;
    float v = acc[r] + bv;
    if (ACT == 1) v = fmaxf(v, 0.f);
    C[rr * (long)N + n] = v;
  }
}

// ---------------------------------------------------------------------------
// Fused GATv2 attention logits:
//   logit[e,h] = sum_c leaky( (e_attr@we)[c] + xl[src][c] + xr[dst][c] ) * att[c]
// ee tile computed by WMMA and never materialized to memory.
// One wave per 16-edge block; cols of one 16-wide tile map to lanes.
// ---------------------------------------------------------------------------
template <int FE>
__global__ __launch_bounds__(256) void gat_logits_k(
    const float* __restrict__ ea, const float* __restrict__ we,
    const float* __restrict__ xl, const float* __restrict__ xr,
    const int* __restrict__ srcf, const int* __restrict__ dstf,
    const float* __restrict__ att, float* __restrict__ logits, int EF, int HC,
    int CH) {
  const int lane = threadIdx.x & 31;
  const int wv = threadIdx.x >> 5;
  const int eb = blockIdx.x * 8 + wv;
  const int e0 = eb * 16;
  if (e0 >= EF) return;                     // wave-uniform
  const int half = lane >> 4;
  const int m = lane & 15;
  const long edge = e0 + m;

  // preload this wave's 16xFE A block as WMMA fragments
  v2f af[FE / 4];
#pragma unroll
  for (int k0 = 0; k0 < FE; k0 += 4)
    af[k0 >> 2] = *(const v2f*)(ea + edge * FE + k0 + 2 * half);

  int sid[8], did[8];
#pragma unroll
  for (int r = 0; r < 8; ++r) {
    int mm = e0 + half * 8 + r;
    sid[r] = srcf[mm];
    did[r] = dstf[mm];
  }

  for (int h = 0; h < 4; ++h) {
    float logacc[8];
#pragma unroll
    for (int r = 0; r < 8; ++r) logacc[r] = 0.f;
    for (int nt = 0; nt < CH; nt += 16) {
      const int col = h * CH + nt + m;
      v8f acc = {0.f, 0.f, 0.f, 0.f, 0.f, 0.f, 0.f, 0.f};
#pragma unroll
      for (int k0 = 0; k0 < FE; k0 += 4) {
        v2f b;
        b.x = we[(long)(k0 + 2 * half) * HC + col];
        b.y = we[(long)(k0 + 2 * half + 1) * HC + col];
        acc = wmma_f32(af[k0 >> 2], b, acc);
      }
      const float attv = att[col];
#pragma unroll
      for (int r = 0; r < 8; ++r) {
        float z = acc[r] + xl[(long)sid[r] * HC + col] +
                  xr[(long)did[r] * HC + col];
        z = (z > 0.f) ? z : 0.2f * z;
        logacc[r] += z * attv;
      }
    }
#pragma unroll
    for (int r = 0; r < 8; ++r) {
      float v = logacc[r];
      v += __shfl_xor(v, 1, 32);
      v += __shfl_xor(v, 2, 32);
      v += __shfl_xor(v, 4, 32);
      v += __shfl_xor(v, 8, 32);            // sum over 16 cols in this half
      if (m == r) logits[(long)(e0 + half * 8 + r) * 4 + h] = v;
    }
  }
}

// ---------------------------------------------------------------------------
// Small supporting kernels
// ---------------------------------------------------------------------------
__global__ void build_srcdst_k(const int* __restrict__ ei, int* __restrict__ s,
                               int* __restrict__ d, int E, int EF) {
  int i = blockIdx.x * blockDim.x + threadIdx.x;
  if (i >= EF) return;
  if (i < E) {
    s[i] = ei[i];
    d[i] = ei[E + i];
  } else {
    s[i] = i - E;
    d[i] = i - E;
  }
}

__global__ void deg_easum_k(const int* __restrict__ dst,
                            const float* __restrict__ ea,
                            float* __restrict__ deg, float* __restrict__ easum,
                            int FE) {
  int e = blockIdx.x;
  int d = dst[e];
  if (threadIdx.x == 0) atomicAdd(deg + d, 1.f);
  for (int c = threadIdx.x; c < FE; c += blockDim.x)
    atomicAdd(easum + (long)d * FE + c, ea[(long)e * FE + c]);
}

__global__ void fill_self_k(float* __restrict__ ea_self,
                            const float* __restrict__ easum,
                            const float* __restrict__ deg, int FE) {
  int n = blockIdx.x;
  float dg = fmaxf(deg[n], 1.f);
  for (int c = threadIdx.x; c < FE; c += blockDim.x)
    ea_self[(long)n * FE + c] = easum[(long)n * FE + c] / dg;
}

__global__ void combine_relu_k(float* __restrict__ t,
                               const float* __restrict__ ps,
                               const float* __restrict__ pd,
                               const int* __restrict__ src,
                               const int* __restrict__ dst, int D) {
  int e = blockIdx.x;
  int s = src[e], d = dst[e];
  for (int c = threadIdx.x; c < D; c += blockDim.x) {
    float v = t[(long)e * D + c] + ps[(long)s * D + c] + pd[(long)d * D + c];
    t[(long)e * D + c] = fmaxf(v, 0.f);
  }
}

__global__ void softmax_max_k(const float* __restrict__ logits,
                              const int* __restrict__ dstf,
                              unsigned* __restrict__ lmaxU, int EF) {
  int i = blockIdx.x * blockDim.x + threadIdx.x;
  if (i >= EF) return;
  int d = dstf[i];
#pragma unroll
  for (int h = 0; h < 4; ++h)
    atomicMax(lmaxU + d * 4 + h, enc_ord(logits[(long)i * 4 + h]));
}

__global__ void softmax_ex_k(float* __restrict__ logits,
                             const int* __restrict__ dstf,
                             const unsigned* __restrict__ lmaxU,
                             float* __restrict__ den, int EF) {
  int i = blockIdx.x * blockDim.x + threadIdx.x;
  if (i >= EF) return;
  int d = dstf[i];
#pragma unroll
  for (int h = 0; h < 4; ++h) {
    float lm = dec_ord(lmaxU[d * 4 + h]);
    float ex = __expf(logits[(long)i * 4 + h] - lm);
    logits[(long)i * 4 + h] = ex;
    atomicAdd(den + d * 4 + h, ex);
  }
}

__global__ __launch_bounds__(256) void aggregate_k(
    const float* __restrict__ ex, const float* __restrict__ den,
    const float* __restrict__ xl, const int* __restrict__ srcf,
    const int* __restrict__ dstf, float* __restrict__ out, int HC, int CH) {
  int e = blockIdx.x;
  __shared__ float alpha[4];
  int s = srcf[e], d = dstf[e];
  if (threadIdx.x < 4)
    alpha[threadIdx.x] =
        ex[(long)e * 4 + threadIdx.x] / den[(long)d * 4 + threadIdx.x];
  __syncthreads();
  for (int c = threadIdx.x; c < HC; c += blockDim.x)
    atomicAdd(out + (long)d * HC + c, alpha[c / CH] * xl[(long)s * HC + c]);
}

__global__ void bias_relu_bn_k(float* __restrict__ a,
                               const float* __restrict__ cb,
                               const float* __restrict__ g,
                               const float* __restrict__ b,
                               const float* __restrict__ mean,
                               const float* __restrict__ var, int HC) {
  int n = blockIdx.x;
  for (int c = threadIdx.x; c < HC; c += blockDim.x) {
    float v = a[(long)n * HC + c] + cb[c];
    v = fmaxf(v, 0.f);
    float s = g[c] * rsqrtf(var[c] + 1e-5f);
    a[(long)n * HC + c] = (v - mean[c]) * s + b[c];
  }
}

__global__ void pool_k(const float* __restrict__ h2,
                       const int* __restrict__ batch,
                       float* __restrict__ pooled, int HC) {
  int n = blockIdx.x;
  int b = batch[n];
  for (int c = threadIdx.x; c < HC; c += blockDim.x)
    atomicAdd(pooled + (long)b * HC + c, h2[(long)n * HC + c]);
}

__global__ __launch_bounds__(64) void fc_head_k(
    const float* __restrict__ pooled, const float* __restrict__ w1,
    const float* __restrict__ b1, const float* __restrict__ w2,
    const float* __restrict__ b2, float* __restrict__ out) {
  int g = blockIdx.x, j = threadIdx.x;
  float acc = b1[j];
  for (int c = 0; c < 256; ++c) acc += pooled[g * 256 + c] * w1[c * 64 + j];
  acc = fmaxf(acc, 0.f) * w2[j];
  __shared__ float red[64];
  red[j] = acc;
  __syncthreads();
  for (int s = 32; s > 0; s >>= 1) {
    if (j < s) red[j] += red[j + s];
    __syncthreads();
  }
  if (j == 0) out[g] = red[0] + b2[0];
}

// ---------------------------------------------------------------------------
// Host-side orchestration
// ---------------------------------------------------------------------------
extern "C" void kernel_launch(void* const* d_in, const int* in_sizes, int n_in,
                              void* d_out, int out_size, void* d_ws,
                              size_t ws_size, hipStream_t stream) {
  (void)in_sizes; (void)n_in; (void)out_size; (void)ws_size;
  const int N = 10000, E = 160000, EF = E + N, G = 64;

  const float* x      = (const float*)d_in[0];
  const int*   ei     = (const int*)d_in[1];
  const float* eattr  = (const float*)d_in[2];
  const int*   batch  = (const int*)d_in[3];
  const float* nt_w1  = (const float*)d_in[4];
  const float* nt_b1  = (const float*)d_in[5];
  const float* nt_w2  = (const float*)d_in[6];
  const float* nt_b2  = (const float*)d_in[7];
  const float* et1_w1 = (const float*)d_in[8];
  const float* et1_b1 = (const float*)d_in[9];
  const float* et1_w2 = (const float*)d_in[10];
  const float* et1_b2 = (const float*)d_in[11];
  const float* c1_wl  = (const float*)d_in[12];
  const float* c1_bl  = (const float*)d_in[13];
  const float* c1_wr  = (const float*)d_in[14];
  const float* c1_br  = (const float*)d_in[15];
  const float* c1_we  = (const float*)d_in[16];
  const float* c1_att = (const float*)d_in[17];
  const float* c1_bias= (const float*)d_in[18];
  const float* bn1_g  = (const float*)d_in[19];
  const float* bn1_b  = (const float*)d_in[20];
  const float* bn1_m  = (const float*)d_in[21];
  const float* bn1_v  = (const float*)d_in[22];
  const float* et2_w1 = (const float*)d_in[23];
  const float* et2_b1 = (const float*)d_in[24];
  const float* et2_w2 = (const float*)d_in[25];
  const float* et2_b2 = (const float*)d_in[26];
  const float* c2_wl  = (const float*)d_in[27];
  const float* c2_bl  = (const float*)d_in[28];
  const float* c2_wr  = (const float*)d_in[29];
  const float* c2_br  = (const float*)d_in[30];
  const float* c2_we  = (const float*)d_in[31];
  const float* c2_att = (const float*)d_in[32];
  const float* c2_bias= (const float*)d_in[33];
  const float* bn2_g  = (const float*)d_in[34];
  const float* bn2_b  = (const float*)d_in[35];
  const float* bn2_m  = (const float*)d_in[36];
  const float* bn2_v  = (const float*)d_in[37];
  const float* fc1_w  = (const float*)d_in[38];
  const float* fc1_b  = (const float*)d_in[39];
  const float* fc2_w  = (const float*)d_in[40];
  const float* fc2_b  = (const float*)d_in[41];

  // workspace arena (static offsets; big regions reused across phases)
  char* ws = (char*)d_ws;
  size_t off = 0;
  auto alloc = [&](size_t bytes) -> char* {
    char* p = ws + off;
    off += (bytes + 255) & ~(size_t)255;
    return p;
  };
  float* h0    = (float*)alloc((size_t)N * 128 * 4);
  float* ea1   = (float*)alloc((size_t)EF * 64 * 4);   // e1 | self-loop attrs
  float* ea2   = (float*)alloc((size_t)EF * 128 * 4);  // e2 | self-loop attrs
  float* big   = (float*)alloc((size_t)2 * N * 1024 * 4);  // xl1|xr1; reused
  float* xl1   = big;
  float* xr1   = big + (size_t)N * 1024;
  float* agg1  = (float*)alloc((size_t)N * 1024 * 4);  // conv1 out -> h1
  int*   srcf  = (int*)alloc((size_t)EF * 4);
  int*   dstf  = (int*)alloc((size_t)EF * 4);
  float* deg   = (float*)alloc((size_t)N * 4);
  float* easum = (float*)alloc((size_t)N * 128 * 4);
  float* logit = (float*)alloc((size_t)EF * 4 * 4);
  unsigned* lmaxU = (unsigned*)alloc((size_t)N * 4 * 4);
  float* den   = (float*)alloc((size_t)N * 4 * 4);
  float* p1    = (float*)alloc((size_t)N * 128 * 4);
  float* p2    = (float*)alloc((size_t)N * 128 * 4);
  float* xl2   = (float*)alloc((size_t)N * 256 * 4);
  float* xr2   = (float*)alloc((size_t)N * 256 * 4);
  float* agg2  = (float*)alloc((size_t)N * 256 * 4);   // conv2 out -> h2
  float* pooled= (float*)alloc((size_t)G * 256 * 4);

  auto gemm = [&](const float* A, const float* B, const float* bias, float* C,
                  int M, int K, int Nc, int act) {
    dim3 grid(M / 16, ((Nc / 16) + 7) / 8);
    if (act)
      gemm_wmma_k<1><<<grid, 256, 0, stream>>>(A, B, bias, C, M, K, Nc);
    else
      gemm_wmma_k<0><<<grid, 256, 0, stream>>>(A, B, bias, C, M, K, Nc);
  };

  // ---- Phase 0: node MLP: h0 = relu(x@w1+b1)@w2+b2 ----
  float* mlp_hid = big;  // [N,256] scratch
  gemm(x, nt_w1, nt_b1, mlp_hid, N, 64, 256, 1);
  gemm(mlp_hid, nt_w2, nt_b2, h0, N, 256, 128, 0);

  // ---- Phase 1: et1 (decomposed concat-GEMM) ----
  gemm(h0, et1_w1 + 16 * 64, nullptr, p1, N, 128, 64, 0);    // src proj
  gemm(h0, et1_w1 + 144 * 64, nullptr, p2, N, 128, 64, 0);   // dst proj
  float* t1 = big;  // [E,64]
  gemm(eattr, et1_w1, et1_b1, t1, E, 16, 64, 0);             // edge proj + b1
  combine_relu_k<<<E, 128, 0, stream>>>(t1, p1, p2, ei, ei + E, 64);
  gemm(t1, et1_w2, et1_b2, ea1, E, 64, 64, 0);               // e1 -> ea1[0:E]

  // ---- Phase 2: GATv2 layer 1 ----
  build_srcdst_k<<<(EF + 255) / 256, 256, 0, stream>>>(ei, srcf, dstf, E, EF);
  (void)hipMemsetAsync(deg, 0, (size_t)N * 4, stream);
  (void)hipMemsetAsync(easum, 0, (size_t)N * 128 * 4, stream);
  deg_easum_k<<<E, 128, 0, stream>>>(ei + E, ea1, deg, easum, 64);
  fill_self_k<<<N, 128, 0, stream>>>(ea1 + (size_t)E * 64, easum, deg, 64);
  gemm(h0, c1_wl, c1_bl, xl1, N, 128, 1024, 0);
  gemm(h0, c1_wr, c1_br, xr1, N, 128, 1024, 0);
  gat_logits_k<64><<<(EF / 16 + 7) / 8, 256, 0, stream>>>(
      ea1, c1_we, xl1, xr1, srcf, dstf, c1_att, logit, EF, 1024, 256);
  (void)hipMemsetAsync(lmaxU, 0, (size_t)N * 16, stream);
  (void)hipMemsetAsync(den, 0, (size_t)N * 16, stream);
  softmax_max_k<<<(EF + 255) / 256, 256, 0, stream>>>(logit, dstf, lmaxU, EF);
  softmax_ex_k<<<(EF + 255) / 256, 256, 0, stream>>>(logit, dstf, lmaxU, den, EF);
  (void)hipMemsetAsync(agg1, 0, (size_t)N * 1024 * 4, stream);
  aggregate_k<<<EF, 256, 0, stream>>>(logit, den, xl1, srcf, dstf, agg1, 1024, 256);
  bias_relu_bn_k<<<N, 256, 0, stream>>>(agg1, c1_bias, bn1_g, bn1_b, bn1_m, bn1_v, 1024);
  float* h1 = agg1;

  // ---- Phase 3: et2 (decomposed concat-GEMM) ----
  gemm(h1, et2_w1 + 64 * 128, nullptr, p1, N, 1024, 128, 0);    // src proj
  gemm(h1, et2_w1 + 1088 * 128, nullptr, p2, N, 1024, 128, 0);  // dst proj
  float* t2 = big;  // [E,128] (xl1/xr1 dead now)
  gemm(ea1, et2_w1, et2_b1, t2, E, 64, 128, 0);                 // e1@W_e + b1
  combine_relu_k<<<E, 128, 0, stream>>>(t2, p1, p2, ei, ei + E, 128);
  gemm(t2, et2_w2, et2_b2, ea2, E, 128, 128, 0);                // e2 -> ea2[0:E]

  // ---- Phase 4: GATv2 layer 2 ----
  (void)hipMemsetAsync(deg, 0, (size_t)N * 4, stream);
  (void)hipMemsetAsync(easum, 0, (size_t)N * 128 * 4, stream);
  deg_easum_k<<<E, 128, 0, stream>>>(ei + E, ea2, deg, easum, 128);
  fill_self_k<<<N, 128, 0, stream>>>(ea2 + (size_t)E * 128, easum, deg, 128);
  gemm(h1, c2_wl, c2_bl, xl2, N, 1024, 256, 0);
  gemm(h1, c2_wr, c2_br, xr2, N, 1024, 256, 0);
  gat_logits_k<128><<<(EF / 16 + 7) / 8, 256, 0, stream>>>(
      ea2, c2_we, xl2, xr2, srcf, dstf, c2_att, logit, EF, 256, 64);
  (void)hipMemsetAsync(lmaxU, 0, (size_t)N * 16, stream);
  (void)hipMemsetAsync(den, 0, (size_t)N * 16, stream);
  softmax_max_k<<<(EF + 255) / 256, 256, 0, stream>>>(logit, dstf, lmaxU, EF);
  softmax_ex_k<<<(EF + 255) / 256, 256, 0, stream>>>(logit, dstf, lmaxU, den, EF);
  (void)hipMemsetAsync(agg2, 0, (size_t)N * 256 * 4, stream);
  aggregate_k<<<EF, 256, 0, stream>>>(logit, den, xl2, srcf, dstf, agg2, 256, 64);
  bias_relu_bn_k<<<N, 256, 0, stream>>>(agg2, c2_bias, bn2_g, bn2_b, bn2_m, bn2_v, 256);
  float* h2 = agg2;

  // ---- Phase 5: pooling + head ----
  (void)hipMemsetAsync(pooled, 0, (size_t)G * 256 * 4, stream);
  pool_k<<<N, 256, 0, stream>>>(h2, batch, pooled, 256);
  fc_head_k<<<G, 64, 0, stream>>>(pooled, fc1_w, fc1_b, fc2_w, fc2_b,
                                  (float*)d_out);
}